// CrossCorrelation_76227079569834
// MI455X (gfx1250) — compile-verified
//
#include <hip/hip_runtime.h>
#include <hip/hip_bf16.h>

#define BB 4
#define NN 64
#define LL 128
#define HH 8
#define EE 32
#define FF 2
#define CC (EE*LL)

typedef __attribute__((ext_vector_type(2))) float v2f;
typedef __attribute__((ext_vector_type(8))) float v8f;

#if defined(__gfx1250__) && \
    __has_builtin(__builtin_amdgcn_global_load_async_to_lds_b128) && \
    __has_builtin(__builtin_amdgcn_s_wait_asynccnt)
#define HAVE_ASYNC_LDS 1
#else
#define HAVE_ASYNC_LDS 0
#endif

#if HAVE_ASYNC_LDS
// Builtin takes AS(1)/AS(3) pointers to a 4 x i32 vector (per hipcc diagnostic).
typedef int async_v4i __attribute__((__vector_size__(16)));
typedef __attribute__((address_space(1))) async_v4i* gv4i_p;
typedef __attribute__((address_space(3))) async_v4i* lv4i_p;
#endif

// ---------------------------------------------------------------------------
// 1) qmean[b,h,e,l] = mean over n of queries[b,n,l,h,e]
// ---------------------------------------------------------------------------
__global__ void qmean_kernel(const float* __restrict__ q, float* __restrict__ qmean) {
  int t = blockIdx.x * blockDim.x + threadIdx.x;   // t < B*H*L*E
  int e = t % EE;
  int l = (t / EE) % LL;
  int h = (t / (EE * LL)) % HH;
  int b = t / (EE * LL * HH);
  const float* p = q + ((((size_t)b * NN) * LL + l) * HH + h) * EE + e;
  float s = 0.f;
#pragma unroll 4
  for (int n = 0; n < NN; ++n) s += p[(size_t)n * (LL * HH * EE)];
  qmean[(((size_t)b * HH + h) * EE + e) * LL + l] = s * (1.0f / NN);
}

// ---------------------------------------------------------------------------
// 2) mean_corr[b,n,h,tau] = (1/E) * sum_{e,s} qmean[b,h,e,(s+tau)%L] * keys[b,n,s,h,e]
//    One workgroup per (b,h): stage the 16KB qmean slice in LDS (async copy),
//    8 waves = 4 n-tiles x 2 tau-halves; each wave holds 4 v8f accumulators
//    and reuses each A fragment (one aligned b64 keys load) across 4 WMMAs.
//    K is enumerated as (s outer, e inner) so A fragments are contiguous.
// ---------------------------------------------------------------------------
__global__ void corr_wmma_kernel(const float* __restrict__ keys,
                                 const float* __restrict__ qmean,
                                 float* __restrict__ mcorr) {
  __shared__ __align__(16) float qs[EE * LL];      // qmean slice [e][l], 16KB

  int bh = blockIdx.x;                  // 0 .. B*H-1
  int b = bh / HH, h = bh % HH;
  const float* qbh = qmean + (((size_t)b * HH + h) * EE) * LL;

  // ---- stage qmean[b,h] into LDS ----
#if HAVE_ASYNC_LDS
  {
#pragma unroll
    for (int it = 0; it < 4; ++it) {
      int idx4 = threadIdx.x + it * 256;           // float4 chunk index (1024 total)
      __builtin_amdgcn_global_load_async_to_lds_b128(
          (gv4i_p)(qbh + idx4 * 4),
          (lv4i_p)(qs + idx4 * 4),
          0, 0);
    }
    __builtin_amdgcn_s_wait_asynccnt(0);
  }
#else
  for (int idx = threadIdx.x; idx < EE * LL; idx += 256) qs[idx] = qbh[idx];
#endif
  __syncthreads();

  int w    = threadIdx.x >> 5;          // wave 0..7
  int lane = threadIdx.x & 31;
  int half = lane >> 4;                 // K-pair selector within a step
  int lm   = lane & 15;
  int ks   = half * 2;

  int nt      = w >> 1;                 // n-tile 0..3
  int tauhalf = w & 1;                  // tau tiles [0..3] or [4..7]
  int t0base  = tauhalf * 64;
  int n       = nt * 16 + lm;           // A-matrix row (node)

  v8f acc[4];
#pragma unroll
  for (int t = 0; t < 4; ++t) acc[t] = (v8f){0.f,0.f,0.f,0.f,0.f,0.f,0.f,0.f};

  const float* krow0 = keys + ((((size_t)b * NN + n) * LL) * HH + h) * EE; // + s*H*E + e

  for (int s = 0; s < LL; ++s) {
    const float* kp = krow0 + (size_t)s * (HH * EE);
    int idx0 = (s + t0base + lm) & (LL - 1);       // B column index for tile 0
#pragma unroll 2
    for (int e0 = 0; e0 < EE; e0 += 4) {
      // A fragment: K = e0+ks, e0+ks+1 at fixed s (contiguous -> one b64 load)
      v2f a = *(const v2f*)(kp + e0 + ks);
      const float* qrow = qs + (e0 + ks) * LL;     // row e0+ks; next K row at +LL
#pragma unroll
      for (int t = 0; t < 4; ++t) {
        int idx = (idx0 + 16 * t) & (LL - 1);
        v2f bb;
        bb.x = qrow[idx];                          // K = e0+ks
        bb.y = qrow[LL + idx];                     // K = e0+ks+1
        acc[t] = __builtin_amdgcn_wmma_f32_16x16x4_f32(
            false, a, false, bb, (short)0, acc[t], false, false);
      }
    }
  }

#pragma unroll
  for (int t = 0; t < 4; ++t) {
    int t0 = t0base + t * 16;
#pragma unroll
    for (int r = 0; r < 8; ++r) {
      int m = r + half * 8;   // D layout: lanes 0-15 -> M=r, lanes 16-31 -> M=r+8
      mcorr[(((size_t)b * NN + (nt * 16 + m)) * HH + h) * LL + t0 + lm] =
          acc[t][r] * (1.0f / EE);
    }
  }
}

// ---------------------------------------------------------------------------
// 3) top-2 over L lags per (b,n,h): delay = L - idx, sigw = sigmoid(value)
// ---------------------------------------------------------------------------
__global__ void topk_kernel(const float* __restrict__ mcorr,
                            int* __restrict__ delay, float* __restrict__ sigw) {
  int t = blockIdx.x * blockDim.x + threadIdx.x;
  if (t >= BB * NN * HH) return;
  int h = t % HH;
  int n = (t / HH) % NN;
  int b = t / (HH * NN);
  const float* row = mcorr + (((size_t)b * NN + n) * HH + h) * LL;
  float v1 = -__builtin_inff(); int i1 = 0;
  for (int l = 0; l < LL; ++l) { float v = row[l]; if (v > v1) { v1 = v; i1 = l; } }
  float v2 = -__builtin_inff(); int i2 = 0;
  for (int l = 0; l < LL; ++l) {
    if (l == i1) continue;
    float v = row[l]; if (v > v2) { v2 = v; i2 = l; }
  }
  int base = ((b * NN + n) * HH + h) * FF;
  delay[base + 0] = LL - i1;
  delay[base + 1] = LL - i2;
  sigw[base + 0] = 1.f / (1.f + __expf(-v1));
  sigw[base + 1] = 1.f / (1.f + __expf(-v2));
}

// ---------------------------------------------------------------------------
// 4) stable argsort of delays over N per (b,h,factor); emit the 1- and 2-back
//    predecessors of each node in sorted order (what the K=3 causal conv needs)
// ---------------------------------------------------------------------------
__global__ void rank_kernel(const int* __restrict__ delay,
                            int* __restrict__ pred1, int* __restrict__ pred2) {
  int bid = blockIdx.x;          // 0 .. B*H*F - 1
  int i  = bid % FF;
  int bh = bid / FF;
  int b = bh / HH, h = bh % HH;
  int n = threadIdx.x;           // 0..63
  __shared__ int d[NN];
  __shared__ int snode[NN];
  d[n] = delay[((b * NN + n) * HH + h) * FF + i];
  __syncthreads();
  int dn = d[n];
  int r = 0;
  for (int m = 0; m < NN; ++m) {
    int dm = d[m];
    r += (dm < dn) || (dm == dn && m < n);   // stable rank
  }
  snode[r] = n;
  __syncthreads();
  int base = (bh * FF + i) * NN;
  pred1[base + n] = (r >= 1) ? snode[r - 1] : -1;
  pred2[base + n] = (r >= 2) ? snode[r - 2] : -1;
}

// ---------------------------------------------------------------------------
// 5) fused: per-factor align -> sigmoid scale -> causal depthwise conv over
//    sorted nodes -> unsort -> align-back -> mean over factors.
//    One block per (b,n,l); 256 threads = (h:8) x (e:32), e fastest (coalesced).
// ---------------------------------------------------------------------------
__global__ void fuse_kernel(const float* __restrict__ values,
                            const float* __restrict__ w_cf,
                            const float* __restrict__ b_cf,
                            const int* __restrict__ delay,
                            const float* __restrict__ sigw,
                            const int* __restrict__ pred1,
                            const int* __restrict__ pred2,
                            float* __restrict__ out) {
  int blk = blockIdx.x;              // B*N*L
  int l = blk % LL;
  int n = (blk / LL) % NN;
  int b = blk / (LL * NN);
  int e = threadIdx.x & (EE - 1);
  int h = threadIdx.x >> 5;
  int bh = b * HH + h;

  float total = 0.f;
#pragma unroll
  for (int i = 0; i < FF; ++i) {
    int dn = delay[((b * NN + n) * HH + h) * FF + i];
    int lp = (l + LL - dn) & (LL - 1);        // align-back index
    int c  = e * LL + lp;                     // conv channel
    float acc = b_cf[c];
    float wv0 = w_cf[c * 3 + 0];
    float wv1 = w_cf[c * 3 + 1];
    float wv2 = w_cf[c * 3 + 2];
    int pbase = (bh * FF + i) * NN + n;
    int m1 = pred1[pbase];
    int m0 = pred2[pbase];
    {   // self tap (sorted position j): aligned index collapses to l
      float sv = sigw[((b * NN + n) * HH + h) * FF + i];
      acc += wv2 * sv * values[((((size_t)b * NN + n) * LL + l) * HH + h) * EE + e];
    }
    if (m1 >= 0) {
      int dm = delay[((b * NN + m1) * HH + h) * FF + i];
      int lv = (lp + dm) & (LL - 1);
      float sv = sigw[((b * NN + m1) * HH + h) * FF + i];
      acc += wv1 * sv * values[((((size_t)b * NN + m1) * LL + lv) * HH + h) * EE + e];
    }
    if (m0 >= 0) {
      int dm = delay[((b * NN + m0) * HH + h) * FF + i];
      int lv = (lp + dm) & (LL - 1);
      float sv = sigw[((b * NN + m0) * HH + h) * FF + i];
      acc += wv0 * sv * values[((((size_t)b * NN + m0) * LL + lv) * HH + h) * EE + e];
    }
    total += acc;
  }
  out[((((size_t)b * NN + n) * LL + l) * HH + h) * EE + e] = total * (1.0f / FF);
}

// ---------------------------------------------------------------------------
extern "C" void kernel_launch(void* const* d_in, const int* in_sizes, int n_in,
                              void* d_out, int out_size, void* d_ws, size_t ws_size,
                              hipStream_t stream) {
  const float* queries = (const float*)d_in[0];
  const float* keys    = (const float*)d_in[1];
  const float* values  = (const float*)d_in[2];
  // d_in[3] = attn_mask (unused by the reference path)
  const float* w_cf    = (const float*)d_in[4];
  const float* b_cf    = (const float*)d_in[5];
  float* out = (float*)d_out;

  // workspace carve-up (all fully overwritten each call)
  float* qmean = (float*)d_ws;                                  // B*H*E*L
  float* mcorr = qmean + (size_t)BB * HH * EE * LL;             // B*N*H*L
  int*   delay = (int*)(mcorr + (size_t)BB * NN * HH * LL);     // B*N*H*F
  float* sigw  = (float*)(delay + (size_t)BB * NN * HH * FF);   // B*N*H*F
  int*   pred1 = (int*)(sigw + (size_t)BB * NN * HH * FF);      // B*H*F*N
  int*   pred2 = pred1 + (size_t)BB * HH * FF * NN;             // B*H*F*N

  qmean_kernel<<<(BB * HH * EE * LL) / 256, 256, 0, stream>>>(queries, qmean);
  corr_wmma_kernel<<<BB * HH, 256, 0, stream>>>(keys, qmean, mcorr);
  topk_kernel<<<(BB * NN * HH + 255) / 256, 256, 0, stream>>>(mcorr, delay, sigw);
  rank_kernel<<<BB * HH * FF, NN, 0, stream>>>(delay, pred1, pred2);
  fuse_kernel<<<BB * NN * LL, HH * EE, 0, stream>>>(values, w_cf, b_cf,
                                                    delay, sigw, pred1, pred2, out);
}